// GATModel_8323646619971
// MI455X (gfx1250) — compile-verified
//
#include <hip/hip_runtime.h>
#include <math.h>

typedef float v2f __attribute__((ext_vector_type(2)));
typedef float v8f __attribute__((ext_vector_type(8)));

#define N_NODES 50000
#define N_EDGES 800000
#define N_GRAPHS 64

// ---- order-preserving float <-> uint encoding for atomicMax on floats ----
__device__ __forceinline__ unsigned enc_f32(float f) {
  unsigned b = __float_as_uint(f);
  return (b & 0x80000000u) ? ~b : (b | 0x80000000u);
}
__device__ __forceinline__ float dec_f32(unsigned u) {
  unsigned b = (u & 0x80000000u) ? (u ^ 0x80000000u) : ~u;
  return __uint_as_float(b);
}

// ======================================================================
// C[M,N] = A[M,K] @ B[K,N] + bias[N]   via V_WMMA_F32_16X16X4_F32.
// Block: 256 threads = 8 waves; block covers 16 rows x 128 cols.
// grid = (N/128, M/16).  K <= 256, multiple of 4. M mult of 16, N of 128.
// A-fragment (16x4 f32): lanes 0-15 = rows, VGPR0/1 = K=k0+2*laneHi (+1).
// B-fragment (4x16 f32): lanes 0-15 = cols, VGPR0/1 = K=k0+2*laneHi (+1).
// D (16x16 f32): VGPR i -> row m0 + i + 8*laneHi, col n0 + laneLo.
// ======================================================================
__global__ __launch_bounds__(256) void gemm_bias_wmma(
    const float* __restrict__ A, const float* __restrict__ B,
    const float* __restrict__ bias, float* __restrict__ C,
    int M, int K, int N) {
  __shared__ float ldsA[16 * 256];
  const int m0 = blockIdx.y * 16;
  const int tid = threadIdx.x;

  // cooperative, coalesced load of the 16 x K A-stripe into LDS
  for (int i = tid; i < 16 * K; i += 256) {
    int r = i / K;
    int c = i - r * K;
    ldsA[i] = A[(m0 + r) * K + c];
  }
  __syncthreads();

  const int wave   = tid >> 5;
  const int lane   = tid & 31;
  const int laneLo = lane & 15;
  const int laneHi = lane >> 4;
  const int n0     = blockIdx.x * 128 + wave * 16;

  const float* __restrict__ ldsRow = &ldsA[laneLo * K];
  v8f acc = {};
  for (int k0 = 0; k0 < K; k0 += 4) {
    const int kk = k0 + 2 * laneHi;
    v2f a, b;
    a.x = ldsRow[kk];
    a.y = ldsRow[kk + 1];
    b.x = B[kk * N + n0 + laneLo];
    b.y = B[(kk + 1) * N + n0 + laneLo];
    acc = __builtin_amdgcn_wmma_f32_16x16x4_f32(
        /*neg_a=*/false, a, /*neg_b=*/false, b,
        /*c_mod=*/(short)0, acc, /*reuse_a=*/false, /*reuse_b=*/false);
  }

  const float bv = bias[n0 + laneLo];
#pragma unroll
  for (int i = 0; i < 8; ++i) {
    const int row = m0 + i + 8 * laneHi;
    C[row * N + n0 + laneLo] = acc[i] + bv;
  }
}

// ======================================================================
// Per-edge attention score: one wave per edge (grid-stride over waves).
//   e_vec = edge_attr[e] @ We   (We kept in LDS, 16 x HC)
//   m     = leaky_relu(xl[src] + xr[dst] + e_vec, 0.2)
//   score[e,h] = dot(m[h,:], att[h,:]);  atomicMax segment max per (dst,h)
// 8 lanes per head (HC/32 channels per lane) -> shfl_xor tree reduce.
// ======================================================================
__global__ __launch_bounds__(256) void gat_score(
    const float* __restrict__ xl, const float* __restrict__ xr,
    const float* __restrict__ edge_attr, const float* __restrict__ We,
    const float* __restrict__ att,
    const int* __restrict__ src, const int* __restrict__ dst,
    float* __restrict__ score, unsigned* __restrict__ smax,
    int HC) {
  __shared__ float sWe[16 * 256];
  __shared__ float sAtt[256];
  const int tid = threadIdx.x;
  for (int i = tid; i < 16 * HC; i += 256) sWe[i] = We[i];
  for (int i = tid; i < HC; i += 256) sAtt[i] = att[i];
  __syncthreads();

  const int lane = tid & 31;
  const int cpl  = HC >> 5;        // channels per lane: 8 (HC=256) or 4 (HC=128)
  const int head = lane >> 3;      // 8 lanes per head in both layers
  const int nwaves = gridDim.x * 8;

  for (int e = blockIdx.x * 8 + (tid >> 5); e < N_EDGES; e += nwaves) {
    const int s = src[e];
    const int d = dst[e];
    float ea[16];
#pragma unroll
    for (int f = 0; f < 16; ++f) ea[f] = edge_attr[e * 16 + f];

    float partial = 0.0f;
    const int base = lane * cpl;
    for (int j = 0; j < cpl; ++j) {
      const int hc = base + j;
      float ev = 0.0f;
#pragma unroll
      for (int f = 0; f < 16; ++f) ev += ea[f] * sWe[f * HC + hc];
      float mv = xl[s * HC + hc] + xr[d * HC + hc] + ev;
      mv = (mv > 0.0f) ? mv : 0.2f * mv;
      partial += mv * sAtt[hc];
    }
    partial += __shfl_xor(partial, 4, 8);
    partial += __shfl_xor(partial, 2, 8);
    partial += __shfl_xor(partial, 1, 8);
    if ((lane & 7) == 0) {
      score[e * 4 + head] = partial;
      atomicMax(&smax[d * 4 + head], enc_f32(partial));
    }
  }
}

// exp(score - segmax) and segment-sum denominator; score overwritten with ex.
__global__ void gat_expsum(const int* __restrict__ dst,
                           float* __restrict__ score,
                           const unsigned* __restrict__ smax,
                           float* __restrict__ den) {
  const int idx = blockIdx.x * blockDim.x + threadIdx.x;
  if (idx >= N_EDGES * 4) return;
  const int e = idx >> 2;
  const int h = idx & 3;
  const int d = dst[e];
  const float mx = dec_f32(smax[d * 4 + h]);
  const float ex = __expf(score[idx] - mx);
  score[idx] = ex;
  atomicAdd(&den[d * 4 + h], ex);
}

// out[dst] += (ex/den) * xl[src]; one wave per edge, lanes over channels.
__global__ __launch_bounds__(256) void gat_aggregate(
    const float* __restrict__ xl, const float* __restrict__ ex,
    const float* __restrict__ den,
    const int* __restrict__ src, const int* __restrict__ dst,
    float* __restrict__ acc, int HC, int logC) {
  const int lane = threadIdx.x & 31;
  const int nwaves = gridDim.x * 8;
  for (int e = blockIdx.x * 8 + (threadIdx.x >> 5); e < N_EDGES; e += nwaves) {
    const int s = src[e];
    const int d = dst[e];
    float w[4];
#pragma unroll
    for (int h = 0; h < 4; ++h)
      w[h] = ex[e * 4 + h] / (den[d * 4 + h] + 1e-16f);
    for (int hc = lane; hc < HC; hc += 32) {
      atomicAdd(&acc[d * HC + hc], w[hc >> logC] * xl[s * HC + hc]);
    }
  }
}

__global__ void bias_relu(float* __restrict__ h, const float* __restrict__ bias,
                          int mask, int total) {
  const int idx = blockIdx.x * blockDim.x + threadIdx.x;
  if (idx >= total) return;
  const float v = h[idx] + bias[idx & mask];
  h[idx] = (v > 0.0f) ? v : 0.0f;
}

// global mean pool (sum + count via atomics)
__global__ void pool_kernel(const float* __restrict__ h, const int* __restrict__ batch,
                            float* __restrict__ pooled, float* __restrict__ cnt) {
  const int idx = blockIdx.x * blockDim.x + threadIdx.x;
  if (idx >= N_NODES * 128) return;
  const int n = idx >> 7;
  const int c = idx & 127;
  const int b = batch[n];
  atomicAdd(&pooled[b * 128 + c], h[idx]);
  if (c == 0) atomicAdd(&cnt[b], 1.0f);
}

__global__ void mlp1_kernel(const float* __restrict__ pooled, const float* __restrict__ cnt,
                            const float* __restrict__ W, const float* __restrict__ b,
                            float* __restrict__ z) {
  const int idx = blockIdx.x * blockDim.x + threadIdx.x;
  if (idx >= N_GRAPHS * 128) return;
  const int g = idx >> 7;
  const int j = idx & 127;
  const float inv = 1.0f / fmaxf(cnt[g], 1.0f);
  float s = 0.0f;
  for (int k = 0; k < 128; ++k) s += pooled[g * 128 + k] * W[k * 128 + j];
  s = s * inv + b[j];
  z[idx] = (s > 0.0f) ? s : 0.0f;
}

__global__ void mlp2_kernel(const float* __restrict__ z, const float* __restrict__ W,
                            const float* __restrict__ b, float* __restrict__ out) {
  const int g = blockIdx.x * blockDim.x + threadIdx.x;
  if (g >= N_GRAPHS) return;
  float s = 0.0f;
  for (int j = 0; j < 128; ++j) s += z[g * 128 + j] * W[j];
  s += b[0];
  out[g] = 1.0f / (1.0f + __expf(-s));
}

extern "C" void kernel_launch(void* const* d_in, const int* in_sizes, int n_in,
                              void* d_out, int out_size, void* d_ws, size_t ws_size,
                              hipStream_t stream) {
  (void)in_sizes; (void)n_in; (void)out_size; (void)ws_size;

  const float* x         = (const float*)d_in[0];
  const float* edge_attr = (const float*)d_in[1];
  const int*   edge_idx  = (const int*)d_in[2];
  const int*   batch     = (const int*)d_in[3];
  const float* Wl0  = (const float*)d_in[4];
  const float* bl0  = (const float*)d_in[5];
  const float* Wr0  = (const float*)d_in[6];
  const float* br0  = (const float*)d_in[7];
  const float* We0  = (const float*)d_in[8];
  const float* att0 = (const float*)d_in[9];
  const float* bias0= (const float*)d_in[10];
  const float* Wl1  = (const float*)d_in[11];
  const float* bl1  = (const float*)d_in[12];
  const float* Wr1  = (const float*)d_in[13];
  const float* br1  = (const float*)d_in[14];
  const float* We1  = (const float*)d_in[15];
  const float* att1 = (const float*)d_in[16];
  const float* bias1= (const float*)d_in[17];
  const float* fc1W = (const float*)d_in[18];
  const float* fc1b = (const float*)d_in[19];
  const float* fc2W = (const float*)d_in[20];
  const float* fc2b = (const float*)d_in[21];
  float* out = (float*)d_out;

  const int* src = edge_idx;
  const int* dst = edge_idx + N_EDGES;

  // ---------------- workspace layout (floats) ----------------
  float* ws = (float*)d_ws;
  float*    xl0    = ws;                         // 50000*256 = 12,800,000
  float*    xr0    = ws + 12800000;              // 12,800,000
  float*    h0     = ws + 25600000;              // 12,800,000 (reused as h1)
  float*    score  = ws + 38400000;              // 800000*4 = 3,200,000
  unsigned* smax   = (unsigned*)(ws + 41600000); // 50000*4 = 200,000
  float*    den    = ws + 41800000;              // 200,000
  float*    pooled = ws + 42000000;              // 64*128 = 8192
  float*    cnt    = ws + 42008192;              // 64
  float*    zbuf   = ws + 42008256;              // 8192
  float*    xl1 = xl0;   // layer-1 overlays (smaller)
  float*    xr1 = xr0;
  float*    h1  = h0;    // h0 no longer needed after layer-1 GEMMs

  const int EDGE_BLOCKS = 4096;  // 8 waves/block, grid-stride over 800K edges

  // ================= Layer 0: in=128 -> H=4,C=64 (HC=256) =================
  {
    dim3 g(256 / 128, N_NODES / 16);  // (2, 3125)
    gemm_bias_wmma<<<g, 256, 0, stream>>>(x, Wl0, bl0, xl0, N_NODES, 128, 256);
    gemm_bias_wmma<<<g, 256, 0, stream>>>(x, Wr0, br0, xr0, N_NODES, 128, 256);
  }
  hipMemsetAsync(smax, 0, (size_t)N_NODES * 4 * 4, stream);
  hipMemsetAsync(den,  0, (size_t)N_NODES * 4 * 4, stream);
  hipMemsetAsync(h0,   0, (size_t)N_NODES * 256 * 4, stream);
  gat_score<<<EDGE_BLOCKS, 256, 0, stream>>>(xl0, xr0, edge_attr, We0, att0,
                                             src, dst, score, smax, 256);
  gat_expsum<<<(N_EDGES * 4 + 255) / 256, 256, 0, stream>>>(dst, score, smax, den);
  gat_aggregate<<<EDGE_BLOCKS, 256, 0, stream>>>(xl0, score, den, src, dst, h0, 256, 6);
  bias_relu<<<(N_NODES * 256 + 255) / 256, 256, 0, stream>>>(h0, bias0, 255, N_NODES * 256);

  // ================= Layer 1: in=256 -> H=4,C=32 (HC=128) =================
  {
    dim3 g(128 / 128, N_NODES / 16);  // (1, 3125)
    gemm_bias_wmma<<<g, 256, 0, stream>>>(h0, Wl1, bl1, xl1, N_NODES, 256, 128);
    gemm_bias_wmma<<<g, 256, 0, stream>>>(h0, Wr1, br1, xr1, N_NODES, 256, 128);
  }
  hipMemsetAsync(smax, 0, (size_t)N_NODES * 4 * 4, stream);
  hipMemsetAsync(den,  0, (size_t)N_NODES * 4 * 4, stream);
  hipMemsetAsync(h1,   0, (size_t)N_NODES * 128 * 4, stream);
  gat_score<<<EDGE_BLOCKS, 256, 0, stream>>>(xl1, xr1, edge_attr, We1, att1,
                                             src, dst, score, smax, 128);
  gat_expsum<<<(N_EDGES * 4 + 255) / 256, 256, 0, stream>>>(dst, score, smax, den);
  gat_aggregate<<<EDGE_BLOCKS, 256, 0, stream>>>(xl1, score, den, src, dst, h1, 128, 5);
  bias_relu<<<(N_NODES * 128 + 255) / 256, 256, 0, stream>>>(h1, bias1, 127, N_NODES * 128);

  // ================= Pool + MLP head =================
  hipMemsetAsync(pooled, 0, (size_t)N_GRAPHS * 128 * 4, stream);
  hipMemsetAsync(cnt,    0, (size_t)N_GRAPHS * 4, stream);
  pool_kernel<<<(N_NODES * 128 + 255) / 256, 256, 0, stream>>>(h1, batch, pooled, cnt);
  mlp1_kernel<<<(N_GRAPHS * 128 + 255) / 256, 256, 0, stream>>>(pooled, cnt, fc1W, fc1b, zbuf);
  mlp2_kernel<<<1, 64, 0, stream>>>(zbuf, fc2W, fc2b, out);
}